// EPGSimulation_89618787598337
// MI455X (gfx1250) — compile-verified
//
#include <hip/hip_runtime.h>

typedef __attribute__((ext_vector_type(2))) float v2f;
typedef __attribute__((ext_vector_type(8))) float v8f;

#define REC_STRIDE 64   // floats per step record (lane-major packed, + zero pad block)

// ---------------------------------------------------------------------------
// Kernel 1: fully parallel precompute of the per-step real affine map
//   s' = M6 * s + K,  s = (Fpr,Fpi,Fmr,Fmi,Zr,Zi)
// folding relaxation (E1,E2), B0 precession (e^{i phi}) and the RF rotation
// (conj() terms expressed as sign patterns on the real 6x6).
//
// Record layout (stride 64 floats), packed so each scan lane reads ONE float4:
//   for m = 0..5:
//     rec[8m+0..3] = { M6[m][0], M6[m][1], M6[m][4], M6[m][5] }   (lanes 0-15)
//     rec[8m+4..7] = { M6[m][2], M6[m][3], K[m],     0        }   (lanes 16-31)
//   rec[48..55]    = 0   (pad rows m>=6 load real zeros -> no masking in scan)
// ---------------------------------------------------------------------------
__global__ void epg_coeff_kernel(const float* __restrict__ fa,
                                 const float* __restrict__ ph,
                                 const float* __restrict__ pT1,
                                 const float* __restrict__ pT2,
                                 const float* __restrict__ pTR,
                                 const float* __restrict__ pB0,
                                 const float* __restrict__ pB1,
                                 float* __restrict__ tab, int n)
{
    int t = blockIdx.x * blockDim.x + threadIdx.x;
    if (t >= n) return;

    float T1 = pT1[0], T2 = pT2[0], TR = pTR[0], B0 = pB0[0], B1 = pB1[0];
    float E1 = expf(-TR / T1);
    float E2 = expf(-TR / T2);
    float R1 = 1.0f - E1;

    float phi = 6.283185307179586f * B0 * TR * 0.001f;
    float ur = E2 * cosf(phi);            // u = E2 * e^{i phi}
    float ui = E2 * sinf(phi);

    float s, c;
    sincosf(0.5f * fa[t] * B1, &s, &c);
    float sb, cb;
    sincosf(ph[t], &sb, &cb);

    float c2 = c * c, s2 = s * s, cs = c * s;
    float c2b = cb * cb - sb * sb;        // Re e^{2i beta}
    float s2b = 2.0f * cb * sb;           // Im e^{2i beta}

    float M00r = c2 * ur, M00i = c2 * ui;                 // c^2 * u
    float e2r = s2 * (c2b * ur - s2b * ui);               // s^2 * e^{2ib} * u
    float e2i = s2 * (c2b * ui + s2b * ur);
    float gr = -cs * sb, gi = cs * cb;                    // g = i*c*s*e^{ib}
    float zfr = gr * ur + gi * ui;                        // conj(g)*u
    float zfi = gr * ui - gi * ur;
    float Mzz = (c2 - s2) * E1;

    float* r = tab + (size_t)t * REC_STRIDE;
    // m=0: Fpr' row [M00r,-M00i, e2r, e2i, E1gr,-E1gi], K0 = R1*gr
    r[0]  = M00r;   r[1]  = -M00i;  r[2]  = E1 * gr;  r[3]  = -E1 * gi;
    r[4]  = e2r;    r[5]  = e2i;    r[6]  = R1 * gr;  r[7]  = 0.0f;
    // m=1: Fpi' row [M00i, M00r, e2i,-e2r, E1gi, E1gr], K1 = R1*gi
    r[8]  = M00i;   r[9]  = M00r;   r[10] = E1 * gi;  r[11] = E1 * gr;
    r[12] = e2i;    r[13] = -e2r;   r[14] = R1 * gi;  r[15] = 0.0f;
    // m=2: Fmr' row [e2r,-e2i, M00r, M00i, E1gr, E1gi], K2 = R1*gr
    r[16] = e2r;    r[17] = -e2i;   r[18] = E1 * gr;  r[19] = E1 * gi;
    r[20] = M00r;   r[21] = M00i;   r[22] = R1 * gr;  r[23] = 0.0f;
    // m=3: Fmi' row [-e2i,-e2r,-M00i, M00r,-E1gi, E1gr], K3 = -R1*gi
    r[24] = -e2i;   r[25] = -e2r;   r[26] = -E1 * gi; r[27] = E1 * gr;
    r[28] = -M00i;  r[29] = M00r;   r[30] = -R1 * gi; r[31] = 0.0f;
    // m=4: Zr' row [zfr,-zfi, zfr, zfi, Mzz, 0], K4 = (c2-s2)*R1
    r[32] = zfr;    r[33] = -zfi;   r[34] = Mzz;      r[35] = 0.0f;
    r[36] = zfr;    r[37] = zfi;    r[38] = (c2 - s2) * R1; r[39] = 0.0f;
    // m=5: Zi' row [zfi, zfr,-zfi, zfr, 0, Mzz], K5 = 0
    r[40] = zfi;    r[41] = zfr;    r[42] = 0.0f;     r[43] = Mzz;
    r[44] = -zfi;   r[45] = zfr;    r[46] = 0.0f;     r[47] = 0.0f;
    // zero pad block for rows m >= 6
    r[48] = 0.0f; r[49] = 0.0f; r[50] = 0.0f; r[51] = 0.0f;
    r[52] = 0.0f; r[53] = 0.0f; r[54] = 0.0f; r[55] = 0.0f;
}

// ---------------------------------------------------------------------------
// Kernel 2: sequential WMMA scan. 16 waves; wave w owns states k in
// [16w,16w+16). Per step: two V_WMMA_F32_16X16X4_F32 compute
//   D = M6[:,0:4]*S[0:4]  +  ( M6[:,4:6]*S[4:6] + K*1 )
// where B1's spare K=2 row is a constant-1 channel carrying the affine K via
// A1's K=2 column, so the C operand is inline zero. EPG shift via
// double-buffered LDS with zero guard slots; one split-barrier per step.
// Next-step coefficients are software-pipelined behind the barrier.
// ---------------------------------------------------------------------------
__global__ __launch_bounds__(512) void epg_scan_kernel(
    const float* __restrict__ tab, float* __restrict__ out, int nsteps)
{
    __shared__ float4 sbuf[2][258];   // [buffer][kIdx = k+1], guards at 0 and 257

    const int tid  = threadIdx.x;
    const int lane = tid & 31;
    const int w    = tid >> 5;
    const int m    = lane & 15;
    const bool lo  = lane < 16;
    const int k    = (w << 4) + m;                       // state index of this lane
    const int offs = (m < 6 ? m * 8 : 48) + (lo ? 0 : 4); // per-lane record offset

    if (tid < 2) {
        sbuf[tid][0]   = make_float4(0.f, 0.f, 0.f, 0.f);
        sbuf[tid][257] = make_float4(0.f, 0.f, 0.f, 0.f);
    }
    __syncthreads();

    // State in WMMA B layout:
    //  Bf vgpr0: lanes0-15 = Fpr(k), lanes16-31 = Fmr(k); vgpr1 = Fpi/Fmi
    //  Bz vgpr0: lanes0-15 = Zr(k),  lanes16-31 = 1.0 (K channel); vgpr1 = Zi/0
    v2f Bf = {0.0f, 0.0f};
    v2f Bz;
    Bz.x = lo ? ((k == 0) ? 1.0f : 0.0f) : 1.0f;   // Z[0]=1; constant-1 channel
    Bz.y = 0.0f;
    const v8f Cz = {0.f, 0.f, 0.f, 0.f, 0.f, 0.f, 0.f, 0.f};

    float4 v = *(const float4*)(tab + offs);        // preload record 0

    int p = 0;
    for (int t = 0; t < nsteps; ++t) {
        v2f A0 = {v.x, v.y};
        v2f A1 = {v.z, v.w};
        v8f D = __builtin_amdgcn_wmma_f32_16x16x4_f32(false, A0, false, Bf,
                                                      (short)0, Cz, false, false);
        D = __builtin_amdgcn_wmma_f32_16x16x4_f32(false, A1, false, Bz,
                                                  (short)0, D, false, false);

        // Software-pipelined load of the next record: its L2 latency hides
        // behind the barrier + LDS exchange below.
        int tn = (t + 1 < nsteps) ? (t + 1) : t;
        v = *(const float4*)(tab + (size_t)tn * REC_STRIDE + offs);
        if (t + 32 < nsteps) {
            __builtin_prefetch((const char*)(tab + (size_t)(t + 32) * REC_STRIDE) + lane * 8, 0, 3);
        }

        // Z does not shift; D rows 4,5 (lanes<16) are already in next-B layout.
        Bz.x = lo ? D[4] : 1.0f;
        Bz.y = lo ? D[5] : 0.0f;

        // EPG shift of Fp (k -> k+1) and Fm (k -> k-1) through LDS.
        if (lo) {
            sbuf[p][k + 1] = make_float4(D[0], D[1], D[2], D[3]);
        }
        __syncthreads();
        // lanes<16 read (Fpr,Fpi) from k-1 (kIdx=k); lanes>=16 read (Fmr,Fmi)
        // from k+1 (kIdx=k+2). Guard slots supply zeros at k=0 / k=255.
        const float* sp = (const float*)&sbuf[p][lo ? k : (k + 2)];
        float2 sv = *(const float2*)(sp + (lo ? 0 : 2));
        Bf.x = sv.x;
        Bf.y = sv.y;

        // Stream the post-shift state (matches the reference scan output):
        // out[t][comp][k], comp = Fpr,Fpi,Fmr,Fmi,Zr,Zi.
        float* ob = out + (size_t)t * 1536 + k;
        __builtin_nontemporal_store(Bf.x, ob + (lo ? 0 : 512));
        __builtin_nontemporal_store(Bf.y, ob + (lo ? 256 : 768));
        if (lo) {
            __builtin_nontemporal_store(Bz.x, ob + 1024);
            __builtin_nontemporal_store(Bz.y, ob + 1280);
        }

        p ^= 1;
    }
}

// ---------------------------------------------------------------------------
extern "C" void kernel_launch(void* const* d_in, const int* in_sizes, int n_in,
                              void* d_out, int out_size, void* d_ws, size_t ws_size,
                              hipStream_t stream)
{
    const float* fa = (const float*)d_in[0];
    const float* ph = (const float*)d_in[1];
    const float* T1 = (const float*)d_in[2];
    const float* T2 = (const float*)d_in[3];
    const float* TR = (const float*)d_in[4];
    // d_in[5] = TE, unused by the math
    const float* B0 = (const float*)d_in[6];
    const float* B1 = (const float*)d_in[7];
    int n = in_sizes[0];
    if (n <= 0) return;

    float* tab = (float*)d_ws;   // n * 64 floats = 4 MB for n=16384

    epg_coeff_kernel<<<(n + 127) / 128, 128, 0, stream>>>(fa, ph, T1, T2, TR, B0, B1, tab, n);
    epg_scan_kernel<<<1, 512, 0, stream>>>(tab, (float*)d_out, n);
}